// GNNAutoencoder_47090021433534
// MI455X (gfx1250) — compile-verified
//
#include <hip/hip_runtime.h>
#include <stdint.h>

typedef __attribute__((ext_vector_type(2))) float v2f;
typedef __attribute__((ext_vector_type(8))) float v8f;

#define NNODES 20000
#define DIN    2000
#define HID1   256
#define HID2   128
#define LATD   64
#define NEDGE  640000
#define EPSF   1e-5f

// Epilogue modes
#define EPI_NONE 0
#define EPI_BIAS 1
#define EPI_BIAS_RELU 2
#define EPI_BIAS_SIGMOID 3

// ---------------------------------------------------------------------------
// elementwise / graph kernels
// ---------------------------------------------------------------------------
__global__ void gnn_fill_kernel(float* __restrict__ p, float v, int n) {
    int i = blockIdx.x * blockDim.x + threadIdx.x;
    if (i < n) p[i] = v;
}

__global__ void gnn_deg_scatter(const long long* __restrict__ dst,
                                float* __restrict__ deg, int e_cnt) {
    int e = blockIdx.x * blockDim.x + threadIdx.x;
    if (e < e_cnt) atomicAdd(&deg[(int)dst[e]], 1.0f);
}

__global__ void gnn_dinv_kernel(float* __restrict__ d, int n) {
    int i = blockIdx.x * blockDim.x + threadIdx.x;
    if (i < n) {
        float v = d[i];
        d[i] = (v > 0.0f) ? rsqrtf(v) : 0.0f;
    }
}

__global__ void gnn_norm_kernel(const long long* __restrict__ src,
                                const long long* __restrict__ dst,
                                const float* __restrict__ dinv,
                                float* __restrict__ nrm, int e_cnt) {
    int e = blockIdx.x * blockDim.x + threadIdx.x;
    if (e < e_cnt) nrm[e] = dinv[(int)src[e]] * dinv[(int)dst[e]];
}

// agg[dst,d] += t[src,d] * norm[e]   (one thread per (edge, feature))
__global__ void gnn_scatter_feat(const long long* __restrict__ src,
                                 const long long* __restrict__ dst,
                                 const float* __restrict__ nrm,
                                 const float* __restrict__ t,
                                 float* __restrict__ agg,
                                 int e_cnt, int d_dim) {
    long long idx = (long long)blockIdx.x * blockDim.x + threadIdx.x;
    long long total = (long long)e_cnt * d_dim;
    if (idx >= total) return;
    int e = (int)(idx / d_dim);
    int d = (int)(idx % d_dim);
    int s = (int)src[e];
    int q = (int)dst[e];
    atomicAdd(&agg[(long long)q * d_dim + d], t[(long long)s * d_dim + d] * nrm[e]);
}

// out = act( agg + t * dinv^2 (self loop) + bias )
__global__ void gnn_finalize(const float* __restrict__ agg,
                             const float* __restrict__ t,
                             const float* __restrict__ dinv,
                             const float* __restrict__ bias,
                             float* __restrict__ out,
                             int n_rows, int d_dim, int do_relu) {
    long long idx = (long long)blockIdx.x * blockDim.x + threadIdx.x;
    long long total = (long long)n_rows * d_dim;
    if (idx >= total) return;
    int i = (int)(idx / d_dim);
    int d = (int)(idx % d_dim);
    float di = dinv[i];
    float v = agg[idx] + t[idx] * di * di + bias[d];
    out[idx] = do_relu ? fmaxf(v, 0.0f) : v;
}

// a = silu( a * (1/sqrt(1+eps)) * g[d] + b[d] )
__global__ void gnn_bn_silu(float* __restrict__ a,
                            const float* __restrict__ g,
                            const float* __restrict__ b,
                            int n_rows, int d_dim) {
    long long idx = (long long)blockIdx.x * blockDim.x + threadIdx.x;
    long long total = (long long)n_rows * d_dim;
    if (idx >= total) return;
    int d = (int)(idx % d_dim);
    float scale = rsqrtf(1.0f + EPSF);
    float v = a[idx] * scale * g[d] + b[d];
    a[idx] = v / (1.0f + expf(-v));   // x * sigmoid(x)
}

__global__ void gnn_leaky(float* __restrict__ a, long long total) {
    long long idx = (long long)blockIdx.x * blockDim.x + threadIdx.x;
    if (idx >= total) return;
    float v = a[idx];
    a[idx] = (v > 0.0f) ? v : 0.2f * v;
}

// wave32 LayerNorm(128) + exact GELU, one wave per row (each lane owns 4 vals)
__global__ void gnn_ln_gelu(float* __restrict__ a,
                            const float* __restrict__ g,
                            const float* __restrict__ b,
                            int n_rows) {
    int wave = threadIdx.x >> 5;
    int lane = threadIdx.x & 31;
    int row = blockIdx.x * (blockDim.x >> 5) + wave;
    if (row >= n_rows) return;
    float x0 = a[(long long)row * 128 + lane * 4 + 0];
    float x1 = a[(long long)row * 128 + lane * 4 + 1];
    float x2 = a[(long long)row * 128 + lane * 4 + 2];
    float x3 = a[(long long)row * 128 + lane * 4 + 3];
    float s = x0 + x1 + x2 + x3;
    #pragma unroll
    for (int o = 16; o > 0; o >>= 1) s += __shfl_xor(s, o, 32);
    float mu = s * (1.0f / 128.0f);
    float d0 = x0 - mu, d1 = x1 - mu, d2 = x2 - mu, d3 = x3 - mu;
    float vs = d0 * d0 + d1 * d1 + d2 * d2 + d3 * d3;
    #pragma unroll
    for (int o = 16; o > 0; o >>= 1) vs += __shfl_xor(vs, o, 32);
    float inv = rsqrtf(vs * (1.0f / 128.0f) + EPSF);
    #pragma unroll
    for (int j = 0; j < 4; ++j) {
        float dv = (j == 0) ? d0 : (j == 1) ? d1 : (j == 2) ? d2 : d3;
        float nv = dv * inv * g[lane * 4 + j] + b[lane * 4 + j];
        float ge = 0.5f * nv * (1.0f + erff(nv * 0.70710678118654752f));
        a[(long long)row * 128 + lane * 4 + j] = ge;
    }
}

// N==1 projection heads: out[i] = act( dot(A[i,:], w) + b0 ), wave per row
__global__ void gnn_dot_head(const float* __restrict__ A,
                             const float* __restrict__ w,
                             const float* __restrict__ b,
                             float* __restrict__ out,
                             int M, int K, int epi) {
    int wave = threadIdx.x >> 5;
    int lane = threadIdx.x & 31;
    int row = blockIdx.x * (blockDim.x >> 5) + wave;
    if (row >= M) return;
    float s = 0.0f;
    for (int k = lane; k < K; k += 32)
        s += A[(long long)row * K + k] * w[k];
    #pragma unroll
    for (int o = 16; o > 0; o >>= 1) s += __shfl_xor(s, o, 32);
    if (lane == 0) {
        float v = s + b[0];
        if (epi == EPI_BIAS_SIGMOID) v = 1.0f / (1.0f + expf(-v));
        out[row] = v;
    }
}

// ---------------------------------------------------------------------------
// f32 WMMA GEMM: C[M,N] = A[M,K] @ B[K,N] (+epilogue).
// Requires M%16==0, K%4==0, N%16==0 (no per-lane guards -> clean inner loop).
// Each wave computes a 16x32 strip (two 16x16 tiles sharing the A fragment).
// B addressing uses pointer bumps (+4N floats / step); the second tile sits
// at a constant +16 floats, folding into the load's immediate offset.
// ---------------------------------------------------------------------------
__global__ void gnn_wmma_gemm_f32(const float* __restrict__ A,
                                  const float* __restrict__ B,
                                  const float* __restrict__ bias,
                                  float* __restrict__ C,
                                  int M, int N, int K, int epi) {
    int lane = threadIdx.x & 31;
    int wave = threadIdx.x >> 5;
    int wpb  = blockDim.x >> 5;
    int mtiles  = M >> 4;
    int ntiles  = N >> 4;
    int npairs  = (ntiles + 1) >> 1;
    int tile = blockIdx.x * wpb + wave;
    if (tile >= mtiles * npairs) return;            // wave-uniform exit
    int mt  = tile / npairs;
    int nt2 = tile - mt * npairs;

    int m  = lane & 15;                 // A: row within tile
    int kb = (lane >> 4) << 1;          // A/B: K sub-offset {0,2} per half-wave
    int n0 = nt2 * 32 + (lane & 15);    // first 16-col tile
    int n1 = n0 + 16;                   // second 16-col tile
    bool has2 = (nt2 * 32 + 16) < N;    // wave-uniform (N%16==0)

    const float* arow = A + (long long)(mt * 16 + m) * K + kb;
    const float* blo  = B + (long long)kb * N + n0;   // row k+kb
    const float* bhi  = blo + N;                      // row k+kb+1
    const long long stepB = 4LL * N;

    v8f acc0 = {};
    v8f acc1 = {};
    if (has2) {
        for (int k = 0; k < K; k += 4) {
            v2f av, bv0, bv1;
            av.x  = arow[k];
            av.y  = arow[k + 1];
            bv0.x = blo[0];
            bv0.y = bhi[0];
            bv1.x = blo[16];               // constant offset -> load immediate
            bv1.y = bhi[16];
            acc0 = __builtin_amdgcn_wmma_f32_16x16x4_f32(
                false, av, false, bv0, (short)0, acc0, false, false);
            acc1 = __builtin_amdgcn_wmma_f32_16x16x4_f32(
                false, av, false, bv1, (short)0, acc1, false, false);
            blo += stepB;
            bhi += stepB;
        }
    } else {
        for (int k = 0; k < K; k += 4) {
            v2f av, bv0;
            av.x  = arow[k];
            av.y  = arow[k + 1];
            bv0.x = blo[0];
            bv0.y = bhi[0];
            acc0 = __builtin_amdgcn_wmma_f32_16x16x4_f32(
                false, av, false, bv0, (short)0, acc0, false, false);
            blo += stepB;
            bhi += stepB;
        }
    }

    float bval0 = (epi != EPI_NONE) ? bias[n0] : 0.0f;
    float bval1 = (epi != EPI_NONE && has2) ? bias[n1] : 0.0f;
    int mb = (lane >> 4) * 8;           // D layout: VGPR v -> row v (lo) / v+8 (hi)
    float* crow = C + (long long)(mt * 16 + mb) * N + n0;
    #pragma unroll
    for (int v = 0; v < 8; ++v) {
        float v0 = acc0[v] + bval0;
        if (epi == EPI_BIAS_RELU)    v0 = fmaxf(v0, 0.0f);
        if (epi == EPI_BIAS_SIGMOID) v0 = 1.0f / (1.0f + expf(-v0));
        crow[0] = v0;
        if (has2) {
            float v1 = acc1[v] + bval1;
            if (epi == EPI_BIAS_RELU)    v1 = fmaxf(v1, 0.0f);
            if (epi == EPI_BIAS_SIGMOID) v1 = 1.0f / (1.0f + expf(-v1));
            crow[16] = v1;                 // constant offset
        }
        crow += N;
    }
}

// ---------------------------------------------------------------------------
// launcher
// ---------------------------------------------------------------------------
static inline int cdivll(long long a, long long b) { return (int)((a + b - 1) / b); }

static void launch_gemm(const float* A, const float* B, const float* bias, float* C,
                        int M, int N, int K, int epi, hipStream_t stream) {
    int ntiles = N >> 4;
    int tiles = (M >> 4) * ((ntiles + 1) >> 1);
    int waves_per_blk = 8;                       // 256 threads
    int blocks = cdivll(tiles, waves_per_blk);
    gnn_wmma_gemm_f32<<<blocks, waves_per_blk * 32, 0, stream>>>(A, B, bias, C, M, N, K, epi);
}

extern "C" void kernel_launch(void* const* d_in, const int* in_sizes, int n_in,
                              void* d_out, int out_size, void* d_ws, size_t ws_size,
                              hipStream_t stream) {
    const float*     x   = (const float*)d_in[0];
    const long long* ei  = (const long long*)d_in[1];
    const float *W1  = (const float*)d_in[2],  *b1  = (const float*)d_in[3];
    const float *W2  = (const float*)d_in[4],  *b2  = (const float*)d_in[5];
    const float *W3  = (const float*)d_in[6],  *b3  = (const float*)d_in[7];
    const float *Wd1 = (const float*)d_in[8],  *bd1 = (const float*)d_in[9];
    const float *Wd2 = (const float*)d_in[10], *bd2 = (const float*)d_in[11];
    const float *Ws1 = (const float*)d_in[12], *bs1 = (const float*)d_in[13];
    const float *bng = (const float*)d_in[14], *bnb = (const float*)d_in[15];
    const float *Ws2 = (const float*)d_in[16], *bs2 = (const float*)d_in[17];
    const float *Wf1 = (const float*)d_in[18], *bf1 = (const float*)d_in[19];
    const float *Wf2 = (const float*)d_in[20], *bf2 = (const float*)d_in[21];
    const float *Wn1 = (const float*)d_in[22], *bn1 = (const float*)d_in[23];
    const float *lng = (const float*)d_in[24], *lnb = (const float*)d_in[25];
    const float *Wn2 = (const float*)d_in[26], *bn2 = (const float*)d_in[27];

    float* out   = (float*)d_out;
    float* recon = out;                                        // [20000,2000]
    float* z     = out + (long long)NNODES * DIN;              // [20000,64]
    float* synth = z   + (long long)NNODES * LATD;             // [20000,1]
    float* flip  = synth + NNODES;                             // [20000,1]
    float* noise = flip  + NNODES;                             // [20000,2000]

    float* ws     = (float*)d_ws;
    float* dinv   = ws;                          // 20000
    float* nrm    = ws + NNODES;                 // 640000
    float* bufT   = nrm + NEDGE;                 // 20000*256 (transform / hidden scratch)
    float* bufAgg = bufT + (long long)NNODES * HID1;   // 20000*256
    float* bufH   = bufAgg + (long long)NNODES * HID1; // 20000*256

    const long long* src = ei;
    const long long* dst = ei + NEDGE;

    const int TB = 256;

    // ---- degree / norm ----
    gnn_fill_kernel<<<cdivll(NNODES, TB), TB, 0, stream>>>(dinv, 1.0f, NNODES); // self-loops
    gnn_deg_scatter<<<cdivll(NEDGE, TB), TB, 0, stream>>>(dst, dinv, NEDGE);
    gnn_dinv_kernel<<<cdivll(NNODES, TB), TB, 0, stream>>>(dinv, NNODES);
    gnn_norm_kernel<<<cdivll(NEDGE, TB), TB, 0, stream>>>(src, dst, dinv, nrm, NEDGE);

    // ---- GCN layer 1: 2000 -> 256, ReLU ----
    launch_gemm(x, W1, nullptr, bufT, NNODES, HID1, DIN, EPI_NONE, stream);
    gnn_fill_kernel<<<cdivll((long long)NNODES * HID1, TB), TB, 0, stream>>>(bufAgg, 0.0f, NNODES * HID1);
    gnn_scatter_feat<<<cdivll((long long)NEDGE * HID1, TB), TB, 0, stream>>>(src, dst, nrm, bufT, bufAgg, NEDGE, HID1);
    gnn_finalize<<<cdivll((long long)NNODES * HID1, TB), TB, 0, stream>>>(bufAgg, bufT, dinv, b1, bufH, NNODES, HID1, 1);

    // ---- GCN layer 2: 256 -> 128, ReLU ----
    launch_gemm(bufH, W2, nullptr, bufT, NNODES, HID2, HID1, EPI_NONE, stream);
    gnn_fill_kernel<<<cdivll((long long)NNODES * HID2, TB), TB, 0, stream>>>(bufAgg, 0.0f, NNODES * HID2);
    gnn_scatter_feat<<<cdivll((long long)NEDGE * HID2, TB), TB, 0, stream>>>(src, dst, nrm, bufT, bufAgg, NEDGE, HID2);
    gnn_finalize<<<cdivll((long long)NNODES * HID2, TB), TB, 0, stream>>>(bufAgg, bufT, dinv, b2, bufH, NNODES, HID2, 1);

    // ---- GCN layer 3: 128 -> 64, linear -> z ----
    launch_gemm(bufH, W3, nullptr, bufT, NNODES, LATD, HID2, EPI_NONE, stream);
    gnn_fill_kernel<<<cdivll((long long)NNODES * LATD, TB), TB, 0, stream>>>(bufAgg, 0.0f, NNODES * LATD);
    gnn_scatter_feat<<<cdivll((long long)NEDGE * LATD, TB), TB, 0, stream>>>(src, dst, nrm, bufT, bufAgg, NEDGE, LATD);
    gnn_finalize<<<cdivll((long long)NNODES * LATD, TB), TB, 0, stream>>>(bufAgg, bufT, dinv, b3, z, NNODES, LATD, 0);

    // ---- recon decoder: relu(z@Wd1+bd1) @ Wd2 + bd2 ----
    launch_gemm(z, Wd1, bd1, bufT, NNODES, HID2, LATD, EPI_BIAS_RELU, stream);
    launch_gemm(bufT, Wd2, bd2, recon, NNODES, DIN, HID2, EPI_BIAS, stream);

    // ---- SyntheticHead: (z@Ws1+bs1)/sqrt(1+eps)*g+b -> silu -> @Ws2+bs2 -> sigmoid ----
    launch_gemm(z, Ws1, bs1, bufT, NNODES, 64, LATD, EPI_BIAS, stream);
    gnn_bn_silu<<<cdivll((long long)NNODES * 64, TB), TB, 0, stream>>>(bufT, bng, bnb, NNODES, 64);
    gnn_dot_head<<<cdivll(NNODES, 8), 256, 0, stream>>>(bufT, Ws2, bs2, synth, NNODES, 64, EPI_BIAS_SIGMOID);

    // ---- LabelFlipHead: leaky(z@Wf1+bf1, 0.2) @ Wf2 + bf2 ----
    launch_gemm(z, Wf1, bf1, bufT, NNODES, 32, LATD, EPI_BIAS, stream);
    gnn_leaky<<<cdivll((long long)NNODES * 32, TB), TB, 0, stream>>>(bufT, (long long)NNODES * 32);
    gnn_dot_head<<<cdivll(NNODES, 8), 256, 0, stream>>>(bufT, Wf2, bf2, flip, NNODES, 32, EPI_BIAS);

    // ---- NoiseHead: LN(z@Wn1+bn1) -> GELU -> @Wn2+bn2 -> sigmoid ----
    launch_gemm(z, Wn1, bn1, bufT, NNODES, HID2, LATD, EPI_BIAS, stream);
    gnn_ln_gelu<<<cdivll(NNODES, 8), 256, 0, stream>>>(bufT, lng, lnb, NNODES);
    launch_gemm(bufT, Wn2, bn2, noise, NNODES, DIN, HID2, EPI_BIAS_SIGMOID, stream);
}